// EmbedAttention_26044681683611
// MI455X (gfx1250) — compile-verified
//
#include <hip/hip_runtime.h>
#include <hip/hip_bf16.h>

// Fused: att = mat @ W  -> mask by len_s -> sparsemax over S, per batch row.
// B=4096, S=128, D=512. Bandwidth-bound (1.07 GB @ 23.3 TB/s ~= 46 us floor).
// GEMV mapped onto V_WMMA_F32_16X16X4_F32 (W in column 0 of B-matrix):
// full fp32 fidelity; 16x compute waste is irrelevant vs the HBM roofline.
// Two independent accumulator chains break the WMMA RAW serialization.

typedef __attribute__((ext_vector_type(2))) float v2f;
typedef __attribute__((ext_vector_type(4))) float v4f;
typedef __attribute__((ext_vector_type(8))) float v8f;

#define SEQ 128
#define DIM 512
#define NEGV (-1e30f)

__global__ __launch_bounds__(256)
void embed_attention_sparsemax(const float* __restrict__ mat,
                               const int*   __restrict__ len_s,
                               const float* __restrict__ W,
                               float*       __restrict__ out)
{
    __shared__ __align__(16) float Wl[DIM];   // 2 KB broadcast copy of W
    __shared__ float att[SEQ];                // masked logits (original order)
    __shared__ float zs[SEQ];                 // sorted (descending)
    __shared__ float cum[SEQ];                // inclusive cumsum of zs
    __shared__ float red[SEQ];                // support-flag reduction
    __shared__ float tau_s;

    const int b    = blockIdx.x;
    const int tid  = threadIdx.x;
    const int wave = tid >> 5;       // 0..7 (wave32)
    const int lane = tid & 31;

    // Stage W into LDS once (512 floats / 256 threads = 2 each).
    for (int i = tid; i < DIM; i += 256) Wl[i] = W[i];
    __syncthreads();

    // ---- Phase 1: GEMV via WMMA. Wave w owns rows s0..s0+15 of this batch.
    const int   s0   = wave * 16;
    const int   m    = lane & 15;          // A-matrix row M = lane % 16
    const int   hi   = lane >> 4;          // lane half selects K pairs
    const float msel = (m == 0) ? 1.0f : 0.0f;   // only column 0 of B is W

    const float* rowp = mat + ((size_t)b * SEQ + (s0 + m)) * DIM;

    v8f acc0 = {};
    v8f acc1 = {};
    for (int kk = 0; kk < DIM; kk += 8) {
        // Stream ahead one full row (2 KB); WGP-scope (locality 3) so it lands
        // in near caches. Tail overrun is speculative -> silently dropped.
        __builtin_prefetch(rowp + kk + DIM, 0, 3);

        // One 16B load per lane feeds two WMMAs:
        //  WMMA#1 covers K = kk+{0,1}(low lanes) / kk+{4,5}(high lanes)
        //  WMMA#2 covers K = kk+{2,3}(low lanes) / kk+{6,7}(high lanes)
        v4f a = *(const v4f*)(rowp + kk + 4 * hi);        // global_load_b128
        v4f w = *(const v4f*)(&Wl[kk + 4 * hi]);          // ds_load_b128 (broadcast)

        v2f a1 = { a.x, a.y };
        v2f a2 = { a.z, a.w };
        v2f b1 = { w.x * msel, w.y * msel };              // B col 0 = W chunk
        v2f b2 = { w.z * msel, w.w * msel };

        // Independent accumulators -> two parallel WMMA dependency chains.
        acc0 = __builtin_amdgcn_wmma_f32_16x16x4_f32(false, a1, false, b1,
                                                     (short)0, acc0, false, false);
        acc1 = __builtin_amdgcn_wmma_f32_16x16x4_f32(false, a2, false, b2,
                                                     (short)0, acc1, false, false);
    }

    // D-matrix column 0: rows 0-7 -> VGPR 0-7 @ lane 0; rows 8-15 @ lane 16.
    const int len = len_s[b];
    if (m == 0) {
        const int base = s0 + 8 * hi;
#pragma unroll
        for (int r = 0; r < 8; ++r) {
            const int s = base + r;
            const float v = acc0[r] + acc1[r];
            att[s] = (s < len) ? v : NEGV;
        }
    }
    __syncthreads();

    // ---- Phase 2: sparsemax over S=128 in LDS (threads 0..127 own one elem).
    if (tid < SEQ) zs[tid] = att[tid];
    __syncthreads();

    // Bitonic sort, descending.
    for (int k = 2; k <= SEQ; k <<= 1) {
        for (int j = k >> 1; j > 0; j >>= 1) {
            if (tid < SEQ) {
                const int ixj = tid ^ j;
                if (ixj > tid) {
                    const float a = zs[tid], c = zs[ixj];
                    const bool up = ((tid & k) == 0);
                    if (up ? (a < c) : (a > c)) { zs[tid] = c; zs[ixj] = a; }
                }
            }
            __syncthreads();
        }
    }

    // Inclusive cumsum (Hillis–Steele).
    if (tid < SEQ) cum[tid] = zs[tid];
    __syncthreads();
    for (int off = 1; off < SEQ; off <<= 1) {
        float v = 0.0f;
        if (tid < SEQ && tid >= off) v = cum[tid - off];
        __syncthreads();
        if (tid < SEQ) cum[tid] += v;
        __syncthreads();
    }

    // Support flags + tree reduction -> k_max.
    if (tid < SEQ)
        red[tid] = ((1.0f + (float)(tid + 1) * zs[tid]) > cum[tid]) ? 1.0f : 0.0f;
    __syncthreads();
    for (int off = SEQ / 2; off > 0; off >>= 1) {
        if (tid < off) red[tid] += red[tid + off];
        __syncthreads();
    }
    if (tid == 0) {
        int kmax = (int)red[0];
        if (kmax < 1) kmax = 1;
        tau_s = (cum[kmax - 1] - 1.0f) / (float)kmax;
    }
    __syncthreads();

    if (tid < SEQ) {
        const float v = att[tid] - tau_s;
        out[(size_t)b * SEQ + tid] = (v > 0.0f) ? v : 0.0f;
    }
}

extern "C" void kernel_launch(void* const* d_in, const int* in_sizes, int n_in,
                              void* d_out, int out_size, void* d_ws, size_t ws_size,
                              hipStream_t stream)
{
    const float* mat  = (const float*)d_in[0];   // [B, S, D] fp32
    const int*   lens = (const int*)  d_in[1];   // [B] int32
    const float* W    = (const float*)d_in[2];   // [D] fp32
    float*       out  = (float*)d_out;           // [B, S, 1] fp32

    const int B = in_sizes[1];                   // 4096
    embed_attention_sparsemax<<<dim3(B), dim3(256), 0, stream>>>(mat, lens, W, out);
}